// _NGCF_24824910971106
// MI455X (gfx1250) — compile-verified
//
#include <hip/hip_runtime.h>
#include <math.h>

#define NUSERS 100000
#define NITEMS 50000
#define NNODES (NUSERS + NITEMS)   // 150000
#define DIM    64
#define NL     3
#define NNZE   4800000
#define NBATCH 16384
#define NTILES (NNODES / 16)       // 9375 (exact)
#define LRELU(x) ((x) > 0.0f ? (x) : 0.2f * (x))

typedef float v2f __attribute__((ext_vector_type(2)));
typedef float v8f __attribute__((ext_vector_type(8)));

// ---------------------------------------------------------------------------
// ego[N,64] = concat(user_emb, item_emb);  all_e[:, 0:64] = ego (layer-0 slab)
// ---------------------------------------------------------------------------
__global__ __launch_bounds__(256) void ngcf_init(const float* __restrict__ ue,
                                                 const float* __restrict__ ie,
                                                 float* __restrict__ ego,
                                                 float* __restrict__ alle) {
  size_t i = (size_t)blockIdx.x * 256 + threadIdx.x;
  if (i >= (size_t)NNODES * DIM) return;
  size_t row = i >> 6;
  int    col = (int)(i & 63);
  float v = (row < (size_t)NUSERS) ? ue[i] : ie[i - (size_t)NUSERS * DIM];
  ego[i] = v;
  alle[row * 256 + col] = v;
}

// ---------------------------------------------------------------------------
// SpMM scatter: side[r,:] += val * ego[c,:]   (one wave per edge, 2 dims/lane)
// ---------------------------------------------------------------------------
__global__ __launch_bounds__(256) void ngcf_spmm(const int* __restrict__ rows,
                                                 const int* __restrict__ cols,
                                                 const float* __restrict__ vals,
                                                 const float* __restrict__ ego,
                                                 float* __restrict__ side) {
  size_t g    = (size_t)blockIdx.x * 256 + threadIdx.x;
  size_t e    = g >> 5;
  int    lane = (int)(g & 31);
  if (e >= (size_t)NNZE) return;
  int   r = rows[e];
  int   c = cols[e];
  float v = vals[e];
  const float2* src = (const float2*)(ego + (size_t)c * DIM);
  float2 x = src[lane];
  float* dst = side + (size_t)r * DIM + 2 * lane;
  unsafeAtomicAdd(dst + 0, v * x.x);   // global_atomic_add_f32
  unsafeAtomicAdd(dst + 1, v * x.y);
}

// ---------------------------------------------------------------------------
// Dense layer (WMMA f32 16x16x4):
//   ego <- lrelu(side@Wgc + bgc) + lrelu((ego*side)@Wbi + bbi)   (in place)
// 256 threads = 8 waves; one 16-row node tile per wave; weights staged in LDS.
// ---------------------------------------------------------------------------
#define WPITCH 65   // pad LDS rows to avoid bank conflicts between lane halves

__global__ __launch_bounds__(256) void ngcf_dense(float* __restrict__ ego,
                                                  const float* __restrict__ side,
                                                  const float* __restrict__ Wgc,
                                                  const float* __restrict__ bgc,
                                                  const float* __restrict__ Wbi,
                                                  const float* __restrict__ bbi) {
  __shared__ float sWgc[64 * WPITCH];
  __shared__ float sWbi[64 * WPITCH];
  __shared__ float sb[128];

  for (int i = threadIdx.x; i < 64 * 64; i += 256) {
    int r = i >> 6, c = i & 63;
    sWgc[r * WPITCH + c] = Wgc[i];
    sWbi[r * WPITCH + c] = Wbi[i];
  }
  if (threadIdx.x < 64) {
    sb[threadIdx.x]      = bgc[threadIdx.x];
    sb[64 + threadIdx.x] = bbi[threadIdx.x];
  }
  __syncthreads();

  int wave = threadIdx.x >> 5;
  int lane = threadIdx.x & 31;
  int tile = blockIdx.x * 8 + wave;
  if (tile >= NTILES) return;

  size_t base = (size_t)tile * 16 * DIM;
  int m  = lane & 15;          // A-frag row
  int ko = (lane >> 4) * 2;    // A-frag K sub-offset (lanes 16-31 hold K+2,K+3)
  int n  = lane & 15;          // B/D-frag column within 16-wide tile

  // Preload all A-fragments (reads of this tile complete before any store ->
  // in-place ego update is safe).
  v2f as_[16];  // side fragment
  v2f ap_[16];  // (ego .* side) fragment
  const float* sp = side + base + (size_t)m * DIM + ko;
  const float* ep = ego  + base + (size_t)m * DIM + ko;
#pragma unroll
  for (int kk = 0; kk < 16; ++kk) {
    float s0 = sp[kk * 4 + 0], s1 = sp[kk * 4 + 1];
    float e0 = ep[kk * 4 + 0], e1 = ep[kk * 4 + 1];
    as_[kk] = (v2f){s0, s1};
    ap_[kk] = (v2f){s0 * e0, s1 * e1};
  }

  int mb = (lane >> 4) * 8;    // D-frag row base (lanes 16-31 hold M=8..15)

#pragma unroll
  for (int ncol = 0; ncol < 4; ++ncol) {
    float bg = sb[ncol * 16 + n];
    float bb = sb[64 + ncol * 16 + n];
    v8f accg = {bg, bg, bg, bg, bg, bg, bg, bg};
    v8f accb = {bb, bb, bb, bb, bb, bb, bb, bb};
#pragma unroll
    for (int kk = 0; kk < 16; ++kk) {
      int kb = kk * 4 + ko;
      v2f bgf = (v2f){sWgc[kb * WPITCH + ncol * 16 + n],
                      sWgc[(kb + 1) * WPITCH + ncol * 16 + n]};
      v2f bbf = (v2f){sWbi[kb * WPITCH + ncol * 16 + n],
                      sWbi[(kb + 1) * WPITCH + ncol * 16 + n]};
      accg = __builtin_amdgcn_wmma_f32_16x16x4_f32(
          false, as_[kk], false, bgf, (short)0, accg, false, false);
      accb = __builtin_amdgcn_wmma_f32_16x16x4_f32(
          false, ap_[kk], false, bbf, (short)0, accb, false, false);
    }
#pragma unroll
    for (int r = 0; r < 8; ++r) {
      float x = accg[r];
      float y = accb[r];
      float v = LRELU(x) + LRELU(y);
      ego[base + (size_t)(mb + r) * DIM + ncol * 16 + n] = v;
    }
  }
}

// ---------------------------------------------------------------------------
// Row L2-normalize ego -> all_e[:, (layer+1)*64 : (layer+2)*64]
// One wave per row; shuffle reduction.
// ---------------------------------------------------------------------------
__global__ __launch_bounds__(256) void ngcf_norm(const float* __restrict__ ego,
                                                 float* __restrict__ alle,
                                                 int layer) {
  size_t g    = (size_t)blockIdx.x * 256 + threadIdx.x;
  size_t row  = g >> 5;
  int    lane = (int)(g & 31);
  if (row >= (size_t)NNODES) return;
  const float2* src = (const float2*)(ego + row * DIM);
  float2 x = src[lane];
  float ss = x.x * x.x + x.y * x.y;
#pragma unroll
  for (int off = 16; off > 0; off >>= 1) ss += __shfl_xor(ss, off, 32);
  float inv = 1.0f / fmaxf(sqrtf(ss), 1e-12f);
  float2* dst = (float2*)(alle + row * 256 + (size_t)(layer + 1) * 64);
  dst[lane] = make_float2(x.x * inv, x.y * inv);
}

// ---------------------------------------------------------------------------
// out[b] = dot(all_e[users[b]], all_e[NUSERS + items[b]]) over 256 dims
// One wave per batch element; float4 loads (8 floats/lane).
// ---------------------------------------------------------------------------
__global__ __launch_bounds__(256) void ngcf_dot(const int* __restrict__ users,
                                                const int* __restrict__ items,
                                                const float* __restrict__ alle,
                                                float* __restrict__ out) {
  size_t g    = (size_t)blockIdx.x * 256 + threadIdx.x;
  size_t b    = g >> 5;
  int    lane = (int)(g & 31);
  if (b >= (size_t)NBATCH) return;
  int u = users[b];
  int it = items[b];
  const float4* up = (const float4*)(alle + (size_t)u * 256);
  const float4* ip = (const float4*)(alle + (size_t)(NUSERS + it) * 256);
  float4 a0 = up[lane],      b0 = ip[lane];
  float4 a1 = up[lane + 32], b1 = ip[lane + 32];
  float s = a0.x * b0.x + a0.y * b0.y + a0.z * b0.z + a0.w * b0.w +
            a1.x * b1.x + a1.y * b1.y + a1.z * b1.z + a1.w * b1.w;
#pragma unroll
  for (int off = 16; off > 0; off >>= 1) s += __shfl_xor(s, off, 32);
  if (lane == 0) out[b] = s;
}

// ---------------------------------------------------------------------------
extern "C" void kernel_launch(void* const* d_in, const int* in_sizes, int n_in,
                              void* d_out, int out_size, void* d_ws, size_t ws_size,
                              hipStream_t stream) {
  (void)in_sizes; (void)n_in; (void)out_size; (void)ws_size;
  const int*   users    = (const int*)d_in[0];
  const int*   items    = (const int*)d_in[1];
  const int*   adj_rows = (const int*)d_in[2];
  const int*   adj_cols = (const int*)d_in[3];
  const float* adj_vals = (const float*)d_in[4];
  const float* user_emb = (const float*)d_in[5];
  const float* item_emb = (const float*)d_in[6];
  const float* W_gc     = (const float*)d_in[7];  // [3,64,64]
  const float* b_gc     = (const float*)d_in[8];  // [3,64]
  const float* W_bi     = (const float*)d_in[9];
  const float* b_bi     = (const float*)d_in[10];
  float* out = (float*)d_out;

  // Workspace layout (floats): ego [N,64] | side [N,64] | all_e [N,256]
  float* ego  = (float*)d_ws;
  float* side = ego  + (size_t)NNODES * DIM;
  float* alle = side + (size_t)NNODES * DIM;

  ngcf_init<<<(NNODES * DIM) / 256, 256, 0, stream>>>(user_emb, item_emb, ego, alle);

  for (int k = 0; k < NL; ++k) {
    hipMemsetAsync(side, 0, (size_t)NNODES * DIM * sizeof(float), stream);
    ngcf_spmm<<<(size_t)NNZE * 32 / 256, 256, 0, stream>>>(adj_rows, adj_cols,
                                                           adj_vals, ego, side);
    ngcf_dense<<<(NTILES + 7) / 8, 256, 0, stream>>>(
        ego, side, W_gc + (size_t)k * 64 * 64, b_gc + (size_t)k * 64,
        W_bi + (size_t)k * 64 * 64, b_bi + (size_t)k * 64);
    ngcf_norm<<<(NNODES * 32) / 256, 256, 0, stream>>>(ego, alle, k);
  }

  ngcf_dot<<<(NBATCH * 32) / 256, 256, 0, stream>>>(users, items, alle, out);
}